// LinearUnit_65867618452250
// MI455X (gfx1250) — compile-verified
//
#include <hip/hip_runtime.h>

// LRU single step: out[b,u] = state[b,u]*a_diag[u] + (x[b,0]+x[b,1])*b_col[u]
// a_diag = concat(as_real[:S], as_imag[:S]); b_col = concat(bs_real[:S], bs_imag[:S])
// d_out = two concatenated copies of new_state (reference returns a tuple).
//
// Bandwidth-bound (384 MB traffic @ 23.3 TB/s => ~16.5 us floor):
//  - non-temporal loads of state / stores of out (don't pollute the 192 MB L2)
//  - rank-1 term via V_WMMA_F32_16X16X4_F32 (only K=0 populated -> exact)
//  - compile-time U so row strides fold into the 24-bit instruction offset:
//    one 32-bit voffset + SGPR base per buffer instead of 64-bit VGPR addr pairs.

typedef __attribute__((ext_vector_type(2))) float v2f;
typedef __attribute__((ext_vector_type(8))) float v8f;

template <int UC>  // UC > 0: compile-time U; UC == -1: runtime U
__global__ __launch_bounds__(256) void lru_step_wmma(
    const float* __restrict__ inputs,    // (B, 2)
    const float* __restrict__ state,     // (B, U)
    const float* __restrict__ as_real,   // (U)
    const float* __restrict__ as_imag,   // (U)
    const float* __restrict__ bs_real,   // (U)
    const float* __restrict__ bs_imag,   // (U)
    float*       __restrict__ out0,      // copy 0: (B, U)
    float*       __restrict__ out1,      // copy 1: (B, U)
    int B, int Urt)
{
    const int U    = (UC > 0) ? UC : Urt;   // constexpr in the fast path
    const int S    = U >> 1;
    const int lane = threadIdx.x & 31;
    const int hi   = lane >> 4;     // 0: lanes 0-15, 1: lanes 16-31
    const int col  = lane & 15;     // N (and A's M) index within the tile

    const int wavesPerBlock = blockDim.x >> 5;
    const int waveId   = blockIdx.x * wavesPerBlock + (int)(threadIdx.x >> 5);
    const int numWaves = gridDim.x * wavesPerBlock;

    const int nTiles     = U >> 4;
    const int totalTiles = (B >> 4) * nTiles;

    // Incremental tile decomposition: two divs before the loop, none inside.
    int t  = waveId;
    int mt = t / nTiles;
    int nt = t - mt * nTiles;
    const int dmt = numWaves / nTiles;
    const int dnt = numWaves - dmt * nTiles;

    for (; t < totalTiles; t += numWaves) {
        const int m0 = mt << 4;
        const int n0 = nt << 4;
        const int n  = n0 + col;

        // Gather diagonal / B-column values; address-select (no exec divergence).
        const float a_val = *((n < S) ? (as_real + n) : (as_imag + (n - S)));
        const float b_val = *((n < S) ? (bs_real + n) : (bs_imag + (n - S)));

        // x_sum for the 16 rows of this tile, one per lane (A-matrix M index).
        const float2 xv = ((const float2*)inputs)[m0 + col];
        const float  xs = xv.x + xv.y;

        // A (16x4 f32): VGPR0 lanes 0-15 = (M=lane, K=0) = x_sum; all else 0.
        v2f Amat; Amat.x = hi ? 0.0f : xs;    Amat.y = 0.0f;
        // B (4x16 f32): VGPR0 lanes 0-15 = (K=0, N=lane) = b_col; all else 0.
        v2f Bmat; Bmat.x = hi ? 0.0f : b_val; Bmat.y = 0.0f;

        // C = state * a_diag in the WMMA C layout: VGPR r holds row
        // m0 + r + 8*hi, column n. With constexpr U, r*U folds into IOFFSET
        // (max 7*8192*4 = 229376 bytes < 2^23), so all 8 loads share one
        // 32-bit voffset (rowBase) against the state SGPR base.
        const int rowBase = (m0 + (hi << 3)) * U + n;
        v8f C;
        #pragma unroll
        for (int r = 0; r < 8; ++r) {
            const float s = __builtin_nontemporal_load(state + rowBase + r * U);
            C[r] = s * a_val;
        }

#if defined(__has_builtin) && __has_builtin(__builtin_amdgcn_wmma_f32_16x16x4_f32)
        // D = A x B + C : exact rank-1 update fused with the diag-scaled tile.
        const v8f D = __builtin_amdgcn_wmma_f32_16x16x4_f32(
            /*neg_a=*/false, Amat, /*neg_b=*/false, Bmat,
            /*c_mod=*/(short)0, C, /*reuse_a=*/false, /*reuse_b=*/false);
#else
        // Exact VALU fallback (also taken on the host Sema pass).
        v8f D;
        #pragma unroll
        for (int r = 0; r < 8; ++r) {
            const float2 rv = ((const float2*)inputs)[m0 + (hi << 3) + r];
            D[r] = __builtin_fmaf(rv.x + rv.y, b_val, C[r]);
        }
#endif

        // Store both tuple copies straight from registers, streaming (NT).
        // Separate SGPR bases for out0/out1 + shared voffset + imm offsets.
        #pragma unroll
        for (int r = 0; r < 8; ++r) {
            __builtin_nontemporal_store(D[r], out0 + rowBase + r * U);
            __builtin_nontemporal_store(D[r], out1 + rowBase + r * U);
        }

        // Advance tile coordinates without div/mod.
        nt += dnt; mt += dmt;
        if (nt >= nTiles) { nt -= nTiles; ++mt; }
    }
}

extern "C" void kernel_launch(void* const* d_in, const int* in_sizes, int n_in,
                              void* d_out, int out_size, void* d_ws, size_t ws_size,
                              hipStream_t stream) {
    const float* inputs  = (const float*)d_in[0];
    const float* state   = (const float*)d_in[1];
    const float* as_real = (const float*)d_in[2];
    const float* as_imag = (const float*)d_in[3];
    const float* bs_real = (const float*)d_in[4];
    const float* bs_imag = (const float*)d_in[5];

    const int B = in_sizes[0] / 2;   // inputs is (B, 2)
    const int U = in_sizes[2];       // NUM_UNITS = 2S

    float* out0 = (float*)d_out;
    float* out1 = out0 + (long long)B * U;   // second tuple copy

    const int totalTiles    = (B / 16) * (U / 16);   // 131072 for default shape
    const int wavesPerBlock = 8;                     // 256 threads = 8 wave32
    const int tilesPerWave  = 4;                     // grid-stride depth
    long long wavesNeeded = (totalTiles + tilesPerWave - 1) / tilesPerWave;
    int blocks = (int)((wavesNeeded + wavesPerBlock - 1) / wavesPerBlock);
    if (blocks < 1) blocks = 1;
    if (blocks > 65535) blocks = 65535;

    if (U == 8192) {
        // Fast path: row stride folds into instruction offsets.
        lru_step_wmma<8192><<<blocks, 256, 0, stream>>>(
            inputs, state, as_real, as_imag, bs_real, bs_imag, out0, out1, B, U);
    } else {
        lru_step_wmma<-1><<<blocks, 256, 0, stream>>>(
            inputs, state, as_real, as_imag, bs_real, bs_imag, out0, out1, B, U);
    }
}